// InvariantExtendedKalmanFilter_8014408974512
// MI455X (gfx1250) — compile-verified
//
#include <hip/hip_runtime.h>
#include <math.h>

typedef __attribute__((ext_vector_type(2))) float v2f;
typedef __attribute__((ext_vector_type(8))) float v8f;

#define LD 33      // padded leading dim (32x32 tiles, stride 33 to dodge bank conflicts)
#define NT 128     // 4 wave32 waves
#define GR 9.80665f

// ---------- tiny 3x3 helpers (run serially on one lane) ----------
__device__ __forceinline__ void skew3(const float* v, float* K) {
  K[0] = 0.f;   K[1] = -v[2]; K[2] =  v[1];
  K[3] =  v[2]; K[4] = 0.f;   K[5] = -v[0];
  K[6] = -v[1]; K[7] =  v[0]; K[8] = 0.f;
}
__device__ __forceinline__ void m3mul(const float* A, const float* B, float* C) {
#pragma unroll
  for (int i = 0; i < 3; i++)
#pragma unroll
    for (int j = 0; j < 3; j++)
      C[i*3+j] = A[i*3]*B[j] + A[i*3+1]*B[3+j] + A[i*3+2]*B[6+j];
}
__device__ __forceinline__ void m3vec(const float* A, const float* x, float* y) {
#pragma unroll
  for (int i = 0; i < 3; i++) y[i] = A[i*3]*x[0] + A[i*3+1]*x[1] + A[i*3+2]*x[2];
}
__device__ __forceinline__ void m3tvec(const float* A, const float* x, float* y) { // A^T x
#pragma unroll
  for (int i = 0; i < 3; i++) y[i] = A[i]*x[0] + A[3+i]*x[1] + A[6+i]*x[2];
}
__device__ __forceinline__ void so3exp3(const float* phi, float* R) {
  float a2 = phi[0]*phi[0] + phi[1]*phi[1] + phi[2]*phi[2];
  float ang = sqrtf(a2);
  float S[9]; skew3(phi, S);
  if (ang < 1e-8f) {
#pragma unroll
    for (int i = 0; i < 9; i++) R[i] = S[i];
    R[0] += 1.f; R[4] += 1.f; R[8] += 1.f;
  } else {
    float inv = 1.f / ang;
    float u[3] = {phi[0]*inv, phi[1]*inv, phi[2]*inv};
    float K[9]; skew3(u, K);
    float K2[9]; m3mul(K, K, K2);
    float s = sinf(ang), c = 1.f - cosf(ang);
#pragma unroll
    for (int i = 0; i < 9; i++) R[i] = s*K[i] + c*K2[i];
    R[0] += 1.f; R[4] += 1.f; R[8] += 1.f;
  }
}
__device__ __forceinline__ void so3ljac3(const float* phi, float* J) {
  float a2 = phi[0]*phi[0] + phi[1]*phi[1] + phi[2]*phi[2];
  float ang = sqrtf(a2);
  float S[9]; skew3(phi, S);
  if (ang < 1e-8f) {
#pragma unroll
    for (int i = 0; i < 9; i++) J[i] = 0.5f*S[i];
    J[0] += 1.f; J[4] += 1.f; J[8] += 1.f;
  } else {
    float S2[9]; m3mul(S, S, S2);
    float c1 = (1.f - cosf(ang)) / a2;
    float c2 = (ang - sinf(ang)) / (a2*ang);
#pragma unroll
    for (int i = 0; i < 9; i++) J[i] = c1*S[i] + c2*S2[i];
    J[0] += 1.f; J[4] += 1.f; J[8] += 1.f;
  }
}

// ---------- 32x32 fp32 matmul via V_WMMA_F32_16X16X4_F32 ----------
// 4 waves, wave w owns output tile (ti,tj)=(w>>1, w&1); 8 chained WMMAs (K=32).
// TB=true computes D = A * B^T via transposed LDS fragment addressing.
// ISA 7.12.2 layouts: A frag lane L -> A[m][k0+koff],A[m][k0+koff+1];
// B frag lane L -> B[k0+koff][n],B[k0+koff+1][n]; D vgpr r -> D[r+8*(L>>4)][n].
template<bool TB>
__device__ __forceinline__ void mm32(const float* Am, const float* Bm, float* Dm) {
  const int lane = threadIdx.x & 31;
  const int wave = threadIdx.x >> 5;
  const int ti = (wave >> 1) & 1;
  const int tj = wave & 1;
  const int mn = lane & 15;
  const int koff = (lane >> 4) << 1;
  const float* arow = Am + (ti*16 + mn)*LD;
  const float* brow = Bm + (tj*16 + mn)*LD;   // used when TB
  v8f acc = {};
#pragma unroll
  for (int kk = 0; kk < 8; ++kk) {
    const int k = kk*4 + koff;
    v2f a, b;
    a.x = arow[k];
    a.y = arow[k+1];
    if (TB) {
      b.x = brow[k];
      b.y = brow[k+1];
    } else {
      b.x = Bm[k*LD + tj*16 + mn];
      b.y = Bm[(k+1)*LD + tj*16 + mn];
    }
    acc = __builtin_amdgcn_wmma_f32_16x16x4_f32(false, a, false, b,
                                                (short)0, acc, false, false);
  }
  const int mh = (lane >> 4) << 3;
#pragma unroll
  for (int r = 0; r < 8; ++r)
    Dm[(ti*16 + mh + r)*LD + tj*16 + mn] = acc[r];
}

// ---------- persistent single-workgroup IEKF scan ----------
__global__ __launch_bounds__(NT, 1)
void iekf_kernel(const float* __restrict__ ts,   const float* __restrict__ gyro,
                 const float* __restrict__ accm, const float* __restrict__ vfwd,
                 const float* __restrict__ mcov,
                 const float* __restrict__ rot0, const float* __restrict__ v0,
                 const float* __restrict__ p0,   const float* __restrict__ P0d,
                 const float* __restrict__ Qd_in,
                 float* __restrict__ outR, float* __restrict__ outV,
                 float* __restrict__ outP, int T)
{
  __shared__ float P[32*LD], Pp[32*LD], F[32*LD], F2[32*LD], Phi[32*LD];
  __shared__ float G[32*LD], Gw[32*LD], T1m[32*LD], T2m[32*LD], IKH[32*LD];
  __shared__ float sRot[9], sRc[9], sV[3], sX[3], sBg[3], sBa[3], sPc[3];
  __shared__ float Rp[9], vp[3], xp[3];
  __shared__ float Hm[3*21], PHt[21*3], Km[21*3], Sinv[9], dxv[21], rr[3];
  __shared__ float Qd[18], shvi[3], shwc[3];
  __shared__ float s_dt, s_w[3], s_a[3], s_vf, s_mc[3];

  const int t = threadIdx.x;

  // one-time init: zero padded matrices (sparse fill positions are static,
  // rewritten every step, so zeros persist), set initial state + outputs.
  for (int e = t; e < 32*LD; e += NT) { P[e]=0.f; F[e]=0.f; G[e]=0.f; Gw[e]=0.f; }
  if (t < 21) P[t*LD + t] = P0d[t];
  if (t < 18) Qd[t] = Qd_in[t];
  if (t < 9)  { sRot[t] = rot0[t]; sRc[t] = (t==0||t==4||t==8) ? 1.f : 0.f; outR[t] = rot0[t]; }
  if (t < 3)  { sV[t]=v0[t]; sX[t]=p0[t]; sBg[t]=0.f; sBa[t]=0.f; sPc[t]=0.f;
                outV[t]=v0[t]; outP[t]=p0[t]; }
  if (t < 63) Hm[t] = 0.f;
  __syncthreads();

  for (int i = 0; i < T-1; ++i) {
    // ---- fetch step inputs, prefetch next ----
    if (t < 3) { s_w[t]=gyro[i*3+t]; s_a[t]=accm[i*3+t]; s_mc[t]=mcov[(i+1)*3+t]; }
    if (t == 0) {
      s_dt = ts[i+1] - ts[i];
      s_vf = vfwd[i+1];
      if (i+2 < T) {
        __builtin_prefetch(&gyro[(i+1)*3], 0, 1);   // global_prefetch_b8
        __builtin_prefetch(&accm[(i+1)*3], 0, 1);
        __builtin_prefetch(&mcov[(i+2)*3], 0, 1);
      }
    }
    __syncthreads();

    // ---- propagation (serial, lane 0) ----
    if (t == 0) {
      float dt = s_dt;
      float phi[3] = {(s_w[0]-sBg[0])*dt, (s_w[1]-sBg[1])*dt, (s_w[2]-sBg[2])*dt};
      float dR[9]; so3exp3(phi, dR);
      float Rn[9]; m3mul(sRot, dR, Rn);
      float am[3] = {s_a[0]-sBa[0], s_a[1]-sBa[1], s_a[2]-sBa[2]};
      float an[3]; m3vec(sRot, am, an);
      an[2] -= GR;
#pragma unroll
      for (int q = 0; q < 9; q++) Rp[q] = Rn[q];
#pragma unroll
      for (int q = 0; q < 3; q++) {
        float vq = sV[q] + an[q]*dt;
        vp[q] = vq;
        xp[q] = sX[q] + (sV[q]+vq)*(0.5f*dt);
      }
    }
    __syncthreads();

    // ---- build F*dt, G*dt, Gw = (G*dt)*Q (9 threads, one 3x3 element each) ----
    if (t < 9) {
      const int bi = t/3, bj = t%3;
      const float dt = s_dt;
      float gv[3] = {0.f, 0.f, -GR};
      float Sg[9]; skew3(gv, Sg);
      float sv[9], sx[9]; skew3(sV, sv); skew3(sX, sx);
      float Rij = sRot[t];
      float vR = sv[bi*3]*sRot[bj] + sv[bi*3+1]*sRot[3+bj] + sv[bi*3+2]*sRot[6+bj];
      float xR = sx[bi*3]*sRot[bj] + sx[bi*3+1]*sRot[3+bj] + sx[bi*3+2]*sRot[6+bj];
      float eye = (bi==bj) ? 1.f : 0.f;
      F[(3+bi)*LD + bj]     =  Sg[t]*dt;
      F[(6+bi)*LD + 3+bj]   =  eye*dt;
      F[bi*LD + 9+bj]       = -Rij*dt;
      F[(3+bi)*LD + 9+bj]   = -vR*dt;
      F[(6+bi)*LD + 9+bj]   = -xR*dt;
      F[(3+bi)*LD + 12+bj]  = -Rij*dt;
      G[bi*LD + bj]         =  Rij*dt;
      G[(3+bi)*LD + bj]     =  vR*dt;
      G[(6+bi)*LD + bj]     =  xR*dt;
      G[(3+bi)*LD + 3+bj]   =  Rij*dt;
      G[(9+bi)*LD + 6+bj]   =  eye*dt;
      G[(12+bi)*LD + 9+bj]  =  eye*dt;
      G[(15+bi)*LD + 12+bj] =  eye*dt;
      G[(18+bi)*LD + 15+bj] =  eye*dt;
      Gw[bi*LD + bj]         = Rij*dt*Qd[bj];
      Gw[(3+bi)*LD + bj]     = vR*dt*Qd[bj];
      Gw[(6+bi)*LD + bj]     = xR*dt*Qd[bj];
      Gw[(3+bi)*LD + 3+bj]   = Rij*dt*Qd[3+bj];
      Gw[(9+bi)*LD + 6+bj]   = eye*dt*Qd[6+bj];
      Gw[(12+bi)*LD + 9+bj]  = eye*dt*Qd[9+bj];
      Gw[(15+bi)*LD + 12+bj] = eye*dt*Qd[12+bj];
      Gw[(18+bi)*LD + 15+bj] = eye*dt*Qd[15+bj];
    }
    __syncthreads();

    // ---- covariance propagation: all heavy matmuls on the WMMA pipe ----
    mm32<false>(F, F, F2);      __syncthreads();   // F^2
    mm32<false>(F2, F, T1m);    __syncthreads();   // F^3
    for (int e = t; e < 1024; e += NT) {
      int ii = e >> 5, j = e & 31, idx = ii*LD + j;
      Phi[idx] = ((ii==j)?1.f:0.f) + F[idx] + 0.5f*F2[idx] + (1.f/6.f)*T1m[idx];
    }
    __syncthreads();
    mm32<true>(Gw, G, T2m);     __syncthreads();   // G Q G^T
    for (int e = t; e < 1024; e += NT) {
      int ii = e >> 5, j = e & 31, idx = ii*LD + j;
      T2m[idx] += P[idx];                           // A = P + G Q G^T
    }
    __syncthreads();
    mm32<false>(Phi, T2m, T1m); __syncthreads();   // Phi * A
    mm32<true>(T1m, Phi, Pp);   __syncthreads();   // P_p = Phi * A * Phi^T

    // ---- innovation (lane 0) ----
    if (t == 0) {
      float wc[3] = {s_w[0]-sBg[0], s_w[1]-sBg[1], s_w[2]-sBg[2]};
      float vi[3]; m3tvec(Rp, vp, vi);             // Rot_p^T v_p
      float vb[3]; m3tvec(sRc, vi, vb);            // Rc^T v_imu
      float spc[9]; skew3(sPc, spc);
      float tw[3]; m3vec(spc, wc, tw);
      rr[0] = s_vf - (vb[0]+tw[0]);
      rr[1] =       -(vb[1]+tw[1]);
      rr[2] =       -(vb[2]+tw[2]);
#pragma unroll
      for (int q = 0; q < 3; q++) { shvi[q]=vi[q]; shwc[q]=wc[q]; }
    }
    __syncthreads();

    // ---- H (3x21), 9 threads ----
    if (t < 9) {
      const int l = t/3, j = t%3;
      float rprc = Rp[j*3]*sRc[l] + Rp[j*3+1]*sRc[3+l] + Rp[j*3+2]*sRc[6+l]; // (Rp@Rc)[j][l]
      Hm[l*21 + 3 + j] = rprc;
      float svi[9]; skew3(shvi, svi);
      Hm[l*21 + 9 + j] = sRc[l]*svi[j] + sRc[3+l]*svi[3+j] + sRc[6+l]*svi[6+j];
      float spc[9]; skew3(sPc, spc);
      Hm[l*21 + 15 + j] = -spc[l*3+j];
      float swc[9]; skew3(shwc, swc);
      Hm[l*21 + 18 + j] = -swc[l*3+j];
    }
    __syncthreads();

    // ---- P_p H^T (21x3), 63 threads ----
    if (t < 63) {
      const int ii = t/3, l = t%3;
      float s = 0.f;
      for (int j = 0; j < 21; j++) s += Pp[ii*LD+j]*Hm[l*21+j];
      PHt[t] = s;
    }
    __syncthreads();

    // ---- S = H P_p H^T + Rm; 3x3 inverse (lane 0) ----
    if (t == 0) {
      float S[9];
#pragma unroll
      for (int l = 0; l < 3; l++)
#pragma unroll
        for (int m = 0; m < 3; m++) {
          float s = (l==m) ? s_mc[l] : 0.f;
          for (int j = 0; j < 21; j++) s += Hm[l*21+j]*PHt[j*3+m];
          S[l*3+m] = s;
        }
      float c00 = S[4]*S[8]-S[5]*S[7];
      float c01 = S[5]*S[6]-S[3]*S[8];
      float c02 = S[3]*S[7]-S[4]*S[6];
      float det = S[0]*c00 + S[1]*c01 + S[2]*c02;
      float id  = 1.f/det;
      Sinv[0]=c00*id; Sinv[1]=(S[2]*S[7]-S[1]*S[8])*id; Sinv[2]=(S[1]*S[5]-S[2]*S[4])*id;
      Sinv[3]=c01*id; Sinv[4]=(S[0]*S[8]-S[2]*S[6])*id; Sinv[5]=(S[2]*S[3]-S[0]*S[5])*id;
      Sinv[6]=c02*id; Sinv[7]=(S[1]*S[6]-S[0]*S[7])*id; Sinv[8]=(S[0]*S[4]-S[1]*S[3])*id;
    }
    __syncthreads();

    // ---- K = P_p H^T S^-1 (21x3); dx = K r ----
    if (t < 63) {
      const int ii = t/3, l = t%3;
      Km[t] = PHt[ii*3]*Sinv[l] + PHt[ii*3+1]*Sinv[3+l] + PHt[ii*3+2]*Sinv[6+l];
    }
    __syncthreads();
    if (t < 21) dxv[t] = Km[t*3]*rr[0] + Km[t*3+1]*rr[1] + Km[t*3+2]*rr[2];
    __syncthreads();

    // ---- state update via SE_2(3) exp (lane 0), emit outputs ----
    if (t == 0) {
      float dR[9]; so3exp3(dxv, dR);
      float J[9];  so3ljac3(dxv, J);
      float Rn[9]; m3mul(dR, Rp, Rn);
      float vn[3], xn[3], jv[3], jx[3];
      m3vec(dR, vp, vn); m3vec(dR, xp, xn);
      m3vec(J, dxv+3, jv); m3vec(J, dxv+6, jx);
      float dRc[9]; so3exp3(dxv+15, dRc);
      float Rcn[9]; m3mul(dRc, sRc, Rcn);
#pragma unroll
      for (int q = 0; q < 3; q++) {
        sV[q] = vn[q]+jv[q]; sX[q] = xn[q]+jx[q];
        sBg[q] += dxv[9+q]; sBa[q] += dxv[12+q]; sPc[q] += dxv[18+q];
      }
#pragma unroll
      for (int q = 0; q < 9; q++) { sRot[q] = Rn[q]; sRc[q] = Rcn[q]; }
#pragma unroll
      for (int q = 0; q < 9; q++) outR[(size_t)(i+1)*9+q] = Rn[q];
#pragma unroll
      for (int q = 0; q < 3; q++) { outV[(size_t)(i+1)*3+q] = sV[q];
                                    outP[(size_t)(i+1)*3+q] = sX[q]; }
    }
    __syncthreads();

    // ---- covariance update: IKH = I - K H; P = IKH P_p IKH^T + K Rm K^T ----
    for (int e = t; e < 1024; e += NT) {
      int ii = e >> 5, j = e & 31, idx = ii*LD + j;
      float v = (ii==j) ? 1.f : 0.f;
      if (ii < 21 && j < 21)
        v -= Km[ii*3]*Hm[j] + Km[ii*3+1]*Hm[21+j] + Km[ii*3+2]*Hm[42+j];
      IKH[idx] = v;
    }
    __syncthreads();
    mm32<false>(IKH, Pp, T1m);  __syncthreads();
    mm32<true>(T1m, IKH, T2m);  __syncthreads();
    for (int e = t; e < 1024; e += NT) {
      int ii = e >> 5, j = e & 31;
      float krk = 0.f;
      if (ii < 21 && j < 21)
        krk = Km[ii*3]*s_mc[0]*Km[j*3] + Km[ii*3+1]*s_mc[1]*Km[j*3+1]
            + Km[ii*3+2]*s_mc[2]*Km[j*3+2];
      P[ii*LD+j] = 0.5f*(T2m[ii*LD+j] + T2m[j*LD+ii]) + krk;   // symmetrize + K Rm K^T
    }
    __syncthreads();
  }
}

extern "C" void kernel_launch(void* const* d_in, const int* in_sizes, int n_in,
                              void* d_out, int out_size, void* d_ws, size_t ws_size,
                              hipStream_t stream) {
  const float* ts   = (const float*)d_in[0];
  const float* gyro = (const float*)d_in[1];
  const float* acc  = (const float*)d_in[2];
  const float* vfwd = (const float*)d_in[3];
  const float* mcov = (const float*)d_in[4];
  const float* rot0 = (const float*)d_in[5];
  const float* v0   = (const float*)d_in[6];
  const float* p0   = (const float*)d_in[7];
  const float* P0d  = (const float*)d_in[8];
  const float* Qd   = (const float*)d_in[9];
  const int T = in_sizes[0];
  float* out  = (float*)d_out;
  float* outR = out;
  float* outV = out + (size_t)T*9;
  float* outP = out + (size_t)T*9 + (size_t)T*3;
  hipLaunchKernelGGL(iekf_kernel, dim3(1), dim3(NT), 0, stream,
                     ts, gyro, acc, vfwd, mcov, rot0, v0, p0, P0d, Qd,
                     outR, outV, outP, T);
}